// Model_87333864997430
// MI455X (gfx1250) — compile-verified
//
#include <hip/hip_runtime.h>
#include <hip/hip_bf16.h>
#include <stdint.h>

// Problem constants (static in the reference):
//   bs = 256, num_pools = 2048, pool_len = 16384
//   copy_len = topk(8) * speculative_num_steps(8) = 64
#define BS              256
#define POOL_LEN        16384
#define COPY_LEN        64
#define REQS_PER_BLOCK  8            // one wave32 per request
#define THREADS         (REQS_PER_BLOCK * 32)

// Gather: out[req*64 + j] = (float) req_to_token[req_pool_indices[req]*POOL_LEN
//                                                + seq_lens[req] + j],  j in [0,64)
//
// CDNA5 path: each lane issues two async global->LDS b32 copies (ASYNCcnt),
// drains with s_wait_asynccnt, then reads LDS (DScnt) and stores coalesced.
// b32 granularity keeps alignment legal for odd seq_lens (int rows are only
// guaranteed 4-byte aligned at arbitrary kv_start).
__global__ __launch_bounds__(THREADS)
void spec_token_gather_kernel(const int* __restrict__ req_pool_indices,
                              const int* __restrict__ req_to_token,
                              const int* __restrict__ seq_lens,
                              float* __restrict__ out) {
    __shared__ int lds_buf[REQS_PER_BLOCK * COPY_LEN];   // 2 KB

    const int lane = threadIdx.x & 31;
    const int wave = threadIdx.x >> 5;
    const int req  = blockIdx.x * REQS_PER_BLOCK + wave;

    // Per-wave uniform scalars (compiler keeps these in SGPRs).
    const int pool_idx = req_pool_indices[req];
    const int kv_start = seq_lens[req];

    const int* src = req_to_token + (size_t)pool_idx * POOL_LEN + kv_start;

    // Lane covers elements {lane, lane+32} of the 64-element slice.
    uint64_t g0 = (uint64_t)(uintptr_t)(src + lane);
    uint64_t g1 = (uint64_t)(uintptr_t)(src + lane + 32);

    // LDS byte addresses (low 32 bits of the flat LDS address == LDS offset).
    uint32_t l0 = (uint32_t)(uintptr_t)(&lds_buf[wave * COPY_LEN + lane]);
    uint32_t l1 = (uint32_t)(uintptr_t)(&lds_buf[wave * COPY_LEN + lane + 32]);

    // Async global -> LDS (GV mode: 64-bit VGPR address, VDST = LDS byte addr).
    asm volatile("global_load_async_to_lds_b32 %0, %1, off"
                 :: "v"(l0), "v"(g0) : "memory");
    asm volatile("global_load_async_to_lds_b32 %0, %1, off"
                 :: "v"(l1), "v"(g1) : "memory");

    // Drain this wave's async ops; each wave only reads its own LDS region,
    // so no workgroup barrier is required.
    asm volatile("s_wait_asynccnt 0" ::: "memory");

    int v0 = lds_buf[wave * COPY_LEN + lane];
    int v1 = lds_buf[wave * COPY_LEN + lane + 32];

    float* dst = out + (size_t)req * COPY_LEN;
    dst[lane]      = (float)v0;   // coalesced b32 stores
    dst[lane + 32] = (float)v1;
}

extern "C" void kernel_launch(void* const* d_in, const int* in_sizes, int n_in,
                              void* d_out, int out_size, void* d_ws, size_t ws_size,
                              hipStream_t stream) {
    (void)in_sizes; (void)n_in; (void)out_size; (void)d_ws; (void)ws_size;

    const int* req_pool_indices = (const int*)d_in[0];  // [256]
    const int* req_to_token     = (const int*)d_in[1];  // [2048, 16384]
    const int* seq_lens         = (const int*)d_in[2];  // [256]
    // d_in[3] = topk (8), d_in[4] = speculative_num_steps (8): static, baked in.
    float* out = (float*)d_out;                         // [256 * 64]

    dim3 grid(BS / REQS_PER_BLOCK);   // 32 blocks
    dim3 block(THREADS);              // 256 threads = 8 wave32
    spec_token_gather_kernel<<<grid, block, 0, stream>>>(
        req_pool_indices, req_to_token, seq_lens, out);
}